// VariantGMM_26740466385349
// MI455X (gfx1250) — compile-verified
//
#include <hip/hip_runtime.h>
#include <math.h>

// ---------------------------------------------------------------------------
// Masked-GMM loss for MI455X (gfx1250, wave32).
// Pass A: per-batch stats einsum  S[k,j] = sum_p pred[k,p] * feat_j[p]
//         via V_WMMA_F32_16X16X4_F32. A rows 0-3 = pred, row 4 = ones
//         (gives ROI pixel count n[b] free). B cols j=0..6 are the feature
//         rows {w, w*x_m, w*x_m^2} PRE-COMPUTED into LDS so fragment builds
//         are branch-free (ds_load_b64 + v_cndmask, no exec juggling).
// Pass B: finalize mu, inv(2*var), log-normalizers (tiny).
// Pass C: mix = sum_k pred * exp(lc_k - sum_m d^2 * i2v), loss = -mean(log).
// Traffic: ~2 x 128MB -> ~11us at 23.3 TB/s; working set fits 192MB L2 so
// pass C largely re-reads from L2. FLOPs are trivial; memory path is king.
// ---------------------------------------------------------------------------

#define EPSF     1e-10f
#define BATCH    64
#define KCOMP    4
#define MCHAN    3
#define PIX      65536
#define TILE     1024
#define PAD      4                          // LDS row pad: bank spread + 16B align
#define SLICES_A 16
#define STAGES   (PIX / SLICES_A / TILE)    // 4
#define SLICES_C 32

typedef __attribute__((ext_vector_type(2))) float v2f;
typedef __attribute__((ext_vector_type(4))) float v4f;
typedef __attribute__((ext_vector_type(8))) float v8f;
typedef __attribute__((ext_vector_type(4))) int   v4i;

// workspace layout (float offsets)
#define WS_S    0                                   // [B][K][7]  atomic accum
#define WS_N    (WS_S   + BATCH*KCOMP*7)            // [B]
#define WS_MU   (WS_N   + BATCH)                    // [B][K][M]
#define WS_I2V  (WS_MU  + BATCH*KCOMP*MCHAN)        // [B][K][M]
#define WS_LCP  (WS_I2V + BATCH*KCOMP*MCHAN)        // [B][K][M] scratch
#define WS_LC   (WS_LCP + BATCH*KCOMP*MCHAN)        // [B][K]
#define WS_ZERO_COUNT (BATCH*KCOMP*7 + BATCH)

__global__ void gmm_zero(float* __restrict__ ws, float* __restrict__ out) {
    int i = blockIdx.x * blockDim.x + threadIdx.x;
    if (i < WS_ZERO_COUNT) ws[WS_S + i] = 0.0f;
    if (i == 0) out[0] = 0.0f;
}

// ---------------- Pass A: WMMA statistics einsum ----------------------------
__global__ __launch_bounds__(256)
void gmm_stats_wmma(const float* __restrict__ pred,
                    const float* __restrict__ inp,
                    const int*   __restrict__ heart,
                    float*       __restrict__ ws) {
    __shared__ float lp[KCOMP][TILE + PAD];   // raw pred rows (A matrix rows 0-3)
    __shared__ float lf[8][TILE + PAD];       // feature rows (B cols 0-6) + zero row

    const int b     = blockIdx.x;
    const int slice = blockIdx.y;
    const int tid   = threadIdx.x;
    const int lane  = tid & 31;
    const int wid   = tid >> 5;       // 8 waves
    const int half  = lane >> 4;      // lanes 0-15 -> WMMA-K {0,1}; 16-31 -> {2,3}
    const int r     = lane & 15;      // A row / B column index

    const float* predb = pred  + (size_t)b * KCOMP * PIX;
    const float* inpb  = inp   + (size_t)b * MCHAN * PIX;
    const int*   hb    = heart + (size_t)b * PIX;

    v8f acc = {};   // 16x16 f32 accumulator (rows 0-4 / cols 0-6 useful)

    const int i0 = tid * 4;           // 256 threads x 4 pixels = TILE

    for (int stage = 0; stage < STAGES; ++stage) {
        const int pbase = slice * (STAGES * TILE) + stage * TILE;
        __syncthreads();              // prior compute done before LDS overwrite

        // -------- coalesced b128 staging + feature precompute --------------
        {
            const v4i hv = *(const v4i*)&hb[pbase + i0];
            v4f wv;
            #pragma unroll
            for (int j = 0; j < 4; ++j) wv[j] = (hv[j] == 1) ? 1.0f : 0.0f;
            *(v4f*)&lf[0][i0] = wv;
            #pragma unroll
            for (int m = 0; m < MCHAN; ++m) {
                const v4f xv = *(const v4f*)&inpb[m * PIX + pbase + i0];
                const v4f f1 = wv * xv;
                *(v4f*)&lf[1 + m][i0] = f1;
                *(v4f*)&lf[1 + MCHAN + m][i0] = f1 * xv;
            }
            *(v4f*)&lf[7][i0] = (v4f){0.0f, 0.0f, 0.0f, 0.0f};
            #pragma unroll
            for (int k = 0; k < KCOMP; ++k)
                *(v4f*)&lp[k][i0] = *(const v4f*)&predb[k * PIX + pbase + i0];

            if (stage + 1 < STAGES) {    // global_prefetch_b8 next stage
                const int pn = pbase + TILE + i0;
                __builtin_prefetch(&hb[pn], 0, 1);
                #pragma unroll
                for (int m = 0; m < MCHAN; ++m)
                    __builtin_prefetch(&inpb[m * PIX + pn], 0, 1);
                #pragma unroll
                for (int k = 0; k < KCOMP; ++k)
                    __builtin_prefetch(&predb[k * PIX + pn], 0, 1);
            }
        }
        __syncthreads();

        // -------- branch-free fragment build + WMMA accumulate -------------
        const int qbase = wid * (TILE / 8);          // 128 pixels per wave
        const float aext = (r == KCOMP) ? 1.0f : 0.0f;  // A rows 4..15
        #pragma unroll 4
        for (int q = 0; q < TILE / 8; q += 4) {      // 4 pixels per WMMA
            const int p0 = qbase + q + (half ? 2 : 0);

            const v2f ap = *(const v2f*)&lp[r & 3][p0];  // ds_load_b64
            const v2f fp = *(const v2f*)&lf[r & 7][p0];  // ds_load_b64

            v2f a, bf;
            a.x  = (r < KCOMP) ? ap.x : aext;            // v_cndmask only
            a.y  = (r < KCOMP) ? ap.y : aext;
            bf.x = (r < 8) ? fp.x : 0.0f;
            bf.y = (r < 8) ? fp.y : 0.0f;

            acc = __builtin_amdgcn_wmma_f32_16x16x4_f32(
                false, a, false, bf, (short)0, acc, false, false);
        }
    }

    // C layout: acc[v], lanes 0-15 => M=v, N=lane. Rows 0-3 = S[k][j], row4/col0 = n.
    if (half == 0 && r < 7) {
        float* S = ws + WS_S + (size_t)(b * KCOMP) * 7;
        #pragma unroll
        for (int k = 0; k < KCOMP; ++k)
            atomicAdd(&S[k * 7 + r], acc[k]);
        if (r == 0)
            atomicAdd(&ws[WS_N + b], acc[KCOMP]);
    }
}

// ---------------- Pass B: finalize mu / inv(2 var) / log-normalizers --------
__global__ __launch_bounds__(1024)
void gmm_finalize(float* __restrict__ ws) {
    const int t = threadIdx.x;
    if (t < BATCH * KCOMP * MCHAN) {
        const int m  = t % MCHAN;
        const int bk = t / MCHAN;
        const float* S = ws + WS_S + (size_t)bk * 7;
        const float s0 = S[0];
        const float s  = s0 + EPSF;
        const float s1 = S[1 + m];
        const float s2 = S[1 + MCHAN + m];
        const float mu  = s1 / s;
        const float var = (s2 - 2.0f * mu * s1 + mu * mu * s0) / s + EPSF;
        ws[WS_MU  + t] = mu;
        ws[WS_I2V + t] = 0.5f / var;
        ws[WS_LCP + t] = -0.5f * __logf(6.2831853071795864f * var);
    }
    __syncthreads();
    if (t < BATCH * KCOMP) {
        float lc = 0.0f;
        #pragma unroll
        for (int m = 0; m < MCHAN; ++m) lc += ws[WS_LCP + t * MCHAN + m];
        ws[WS_LC + t] = lc;
    }
}

// ---------------- Pass C: masked negative log-likelihood --------------------
__global__ __launch_bounds__(256)
void gmm_loss(const float* __restrict__ pred,
              const float* __restrict__ inp,
              const int*   __restrict__ heart,
              const float* __restrict__ ws,
              float*       __restrict__ out) {
    const int b     = blockIdx.x;
    const int slice = blockIdx.y;
    const int tid   = threadIdx.x;

    const float* predb = pred  + (size_t)b * KCOMP * PIX;
    const float* inpb  = inp   + (size_t)b * MCHAN * PIX;
    const int*   hb    = heart + (size_t)b * PIX;

    float mu[KCOMP][MCHAN], i2v[KCOMP][MCHAN], lc[KCOMP];
    #pragma unroll
    for (int k = 0; k < KCOMP; ++k) {
        lc[k] = ws[WS_LC + b * KCOMP + k];
        #pragma unroll
        for (int m = 0; m < MCHAN; ++m) {
            mu[k][m]  = ws[WS_MU  + (b * KCOMP + k) * MCHAN + m];
            i2v[k][m] = ws[WS_I2V + (b * KCOMP + k) * MCHAN + m];
        }
    }

    float local = 0.0f;
    const int CHUNK = PIX / SLICES_C;                 // 2048
    const int pbase = slice * CHUNK;
    for (int base = 0; base < CHUNK; base += 4 * 256) {
        const int p = pbase + base + tid * 4;

        const v4i hv = *(const v4i*)&hb[p];
        v4f xm[MCHAN], pr[KCOMP];
        #pragma unroll
        for (int m = 0; m < MCHAN; ++m) xm[m] = *(const v4f*)&inpb[m * PIX + p];
        #pragma unroll
        for (int k = 0; k < KCOMP; ++k) pr[k] = *(const v4f*)&predb[k * PIX + p];

        #pragma unroll
        for (int j = 0; j < 4; ++j) {
            float mix = 0.0f;
            #pragma unroll
            for (int k = 0; k < KCOMP; ++k) {
                float e = lc[k];
                #pragma unroll
                for (int m = 0; m < MCHAN; ++m) {
                    const float d = xm[m][j] - mu[k][m];
                    e -= d * d * i2v[k][m];
                }
                mix += pr[k][j] * __expf(e);
            }
            local += (hv[j] == 1) ? __logf(mix + EPSF) : 0.0f;  // select, no branch
        }
    }

    __shared__ float red[256];
    red[tid] = local;
    __syncthreads();
    #pragma unroll
    for (int s = 128; s > 0; s >>= 1) {
        if (tid < s) red[tid] += red[tid + s];
        __syncthreads();
    }
    if (tid == 0) {
        const float n = ws[WS_N + b];
        atomicAdd(out, -red[0] / (n * (float)BATCH));
    }
}

// ---------------------------------------------------------------------------
extern "C" void kernel_launch(void* const* d_in, const int* in_sizes, int n_in,
                              void* d_out, int out_size, void* d_ws, size_t ws_size,
                              hipStream_t stream) {
    const float* pred  = (const float*)d_in[0];
    const float* inp   = (const float*)d_in[1];
    const int*   heart = (const int*)d_in[2];
    float* out = (float*)d_out;
    float* ws  = (float*)d_ws;

    gmm_zero<<<(WS_ZERO_COUNT + 255) / 256, 256, 0, stream>>>(ws, out);
    gmm_stats_wmma<<<dim3(BATCH, SLICES_A), 256, 0, stream>>>(pred, inp, heart, ws);
    gmm_finalize<<<1, 1024, 0, stream>>>(ws);
    gmm_loss<<<dim3(BATCH, SLICES_C), 256, 0, stream>>>(pred, inp, heart, ws, out);
}